// VectorQuantizer_16252156248457
// MI455X (gfx1250) — compile-verified
//
#include <hip/hip_runtime.h>

#define N_ROWS   65536
#define DIM      64
#define K_CODES  1024
#define BLOCK_THREADS 256
#define WAVES_PER_BLOCK 8
#define ROWS_PER_WAVE 32          // two 16-row WMMA tiles per wave
#define ROWS_PER_BLOCK 256        // 8 waves * 32 rows
#define NUM_BLOCKS 256            // 65536 / 256
#define OUT_ELEMS 4194304
#define TILE_PAD 36               // 32 frag floats + 4 pad (bank spread, 16B mult)

typedef __attribute__((ext_vector_type(2))) float v2f;
typedef __attribute__((ext_vector_type(4))) float v4f;
typedef __attribute__((ext_vector_type(8))) float v8f;

// ---------------------------------------------------------------------------
// Kernel 1: codebook squared norms  enorm[k] = sum_d E[k][d]^2
// ---------------------------------------------------------------------------
__global__ void vq_enorm_kernel(const float* __restrict__ E,
                                float* __restrict__ enorm) {
    int k = blockIdx.x * blockDim.x + threadIdx.x;
    if (k < K_CODES) {
        const float* row = E + k * DIM;
        float s = 0.f;
        #pragma unroll
        for (int i = 0; i < DIM; ++i) { float v = row[i]; s += v * v; }
        enorm[k] = s;
    }
}

// ---------------------------------------------------------------------------
// Kernel 2: fused GEMM (X * E^T via V_WMMA_F32_16X16X4_F32) + argmin + gather.
// Codebook tiles staged through double-buffered LDS in B-fragment layout.
// Four independent WMMA accumulator chains (2 row tiles x 2 K-halves) so each
// WMMA depends on a result 4 instructions back (covers pipeline latency).
// ---------------------------------------------------------------------------
__global__ __launch_bounds__(BLOCK_THREADS) void vq_main_kernel(
    const float* __restrict__ X,      // [65536, 64]
    const float* __restrict__ E,      // [1024, 64]
    const float* __restrict__ enorm,  // [1024]
    float* __restrict__ out,          // [65536, 64] quantized
    float* __restrict__ partials)     // [NUM_BLOCKS] loss partial sums
{
    __shared__ float tile[2][32][TILE_PAD];   // B fragments: [buf][lane][2*s+i]
    __shared__ float s_wave[WAVES_PER_BLOCK];

    const int tid  = threadIdx.x;
    const int lane = tid & 31;
    const int wave = tid >> 5;
    const int half = lane >> 4;               // 0: lanes 0-15, 1: lanes 16-31
    const int l16  = lane & 15;
    const int rowBase = blockIdx.x * ROWS_PER_BLOCK + wave * ROWS_PER_WAVE;

    // Staging role: thread t fills tile[buf][sL][4*sC .. 4*sC+3].
    // tile[L][2s+i] = E[kbase + (L&15)][4s + 2*(L>>4) + i]
    const int sL = tid >> 3;                  // fragment lane 0..31
    const int sC = tid & 7;                   // chunk 0..7 (s pair 2c,2c+1)
    const float* sg = E + (size_t)(sL & 15) * DIM + 8 * sC + 2 * (sL >> 4);

    // ---- A fragments for two 16-row tiles (ISA 32-bit A 16x4 layout:
    // per lane contiguous float2 at column 4*s + 2*half of row rowBase+l16).
    v2f a0[16], a1[16];
    {
        const float* xr0 = X + (size_t)(rowBase + l16) * DIM + 2 * half;
        const float* xr1 = xr0 + 16 * DIM;
        #pragma unroll
        for (int s = 0; s < 16; ++s) {
            a0[s] = *(const v2f*)(xr0 + 4 * s);
            a1[s] = *(const v2f*)(xr1 + 4 * s);
        }
    }

    // Running argmin of (||e||^2 - 2 x.e); C/D layout: VGPR j -> row
    // rowBase + 16*t + j + 8*half, col N = l16 (code kbase + l16).
    float minv[2][8];
    int   mini[2][8];
    #pragma unroll
    for (int t = 0; t < 2; ++t)
        #pragma unroll
        for (int j = 0; j < 8; ++j) { minv[t][j] = 3.4028235e38f; mini[t][j] = 0; }

    // ---- prologue: stage tile 0 into buffer 0
    {
        v2f r0 = *(const v2f*)(sg);
        v2f r1 = *(const v2f*)(sg + 4);
        *(v4f*)&tile[0][sL][4 * sC] = (v4f){r0.x, r0.y, r1.x, r1.y};
    }
    __syncthreads();

    for (int ct = 0; ct < K_CODES / 16; ++ct) {
        const int kbase = ct * 16;
        const int buf = ct & 1;
        const bool hasNext = (ct + 1) < (K_CODES / 16);

        // issue next tile's global loads early (overlap with WMMA work)
        v2f r0 = {}, r1 = {};
        if (hasNext) {
            const float* g = sg + (size_t)(kbase + 16) * DIM;
            r0 = *(const v2f*)(g);
            r1 = *(const v2f*)(g + 4);
        }

        // consume current tile: B fragments from LDS (low latency)
        const float* bp = &tile[buf][lane][0];
        v2f b[16];
        #pragma unroll
        for (int s = 0; s < 16; ++s) b[s] = *(const v2f*)(bp + 2 * s);

        // 4 independent accumulator chains: (row tile 0/1) x (K half a/b)
        v8f c0a = {}, c1a = {}, c0b = {}, c1b = {};
        #pragma unroll
        for (int s = 0; s < 8; ++s) {
            c0a = __builtin_amdgcn_wmma_f32_16x16x4_f32(
                      false, a0[s],     false, b[s],     (short)0, c0a, false, false);
            c1a = __builtin_amdgcn_wmma_f32_16x16x4_f32(
                      false, a1[s],     false, b[s],     (short)0, c1a, false, false);
            c0b = __builtin_amdgcn_wmma_f32_16x16x4_f32(
                      false, a0[s + 8], false, b[s + 8], (short)0, c0b, false, false);
            c1b = __builtin_amdgcn_wmma_f32_16x16x4_f32(
                      false, a1[s + 8], false, b[s + 8], (short)0, c1b, false, false);
        }

        const float en = enorm[kbase + l16];
        #pragma unroll
        for (int j = 0; j < 8; ++j) {
            float d0 = en - 2.0f * (c0a[j] + c0b[j]);
            float d1 = en - 2.0f * (c1a[j] + c1b[j]);
            if (d0 < minv[0][j]) { minv[0][j] = d0; mini[0][j] = kbase + l16; }
            if (d1 < minv[1][j]) { minv[1][j] = d1; mini[1][j] = kbase + l16; }
        }

        // publish next tile after compute; barrier orders producer/consumer
        if (hasNext)
            *(v4f*)&tile[buf ^ 1][sL][4 * sC] = (v4f){r0.x, r0.y, r1.x, r1.y};
        __syncthreads();
    }

    // ---- per-tile cross-lane argmin (butterfly inside each 16-lane half;
    // ties -> lowest code index like jnp.argmin), then gather + loss.
    float acc = 0.f;
    const int c0col = lane * 2;
    #pragma unroll
    for (int t = 0; t < 2; ++t) {
        #pragma unroll
        for (int j = 0; j < 8; ++j) {
            float v = minv[t][j]; int i = mini[t][j];
            #pragma unroll
            for (int m = 1; m <= 8; m <<= 1) {
                float ov = __shfl_xor(v, m, 32);
                int   oi = __shfl_xor(i, m, 32);
                if (ov < v || (ov == v && oi < i)) { v = ov; i = oi; }
            }
            minv[t][j] = v; mini[t][j] = i;
        }
        int oidx[8];
        #pragma unroll
        for (int j = 0; j < 8; ++j) oidx[j] = __shfl_xor(mini[t][j], 16, 32);

        const int tileBase = rowBase + 16 * t;
        #pragma unroll
        for (int rr = 0; rr < 16; ++rr) {
            const int j = rr & 7;
            const int code = ((rr >> 3) == half) ? mini[t][j] : oidx[j];
            const int row = tileBase + rr;
            v2f q = *(const v2f*)(E + (size_t)code * DIM + c0col);
            v2f x = *(const v2f*)(X + (size_t)row  * DIM + c0col);
            *(v2f*)(out + (size_t)row * DIM + c0col) = q;
            float dx = q.x - x.x, dy = q.y - x.y;
            acc += dx * dx + dy * dy;
        }
    }

    // wave reduce (deterministic), fixed-order block combine
    #pragma unroll
    for (int m = 1; m <= 16; m <<= 1) acc += __shfl_xor(acc, m, 32);
    if (lane == 0) s_wave[wave] = acc;
    __syncthreads();
    if (tid == 0) {
        float s = 0.f;
        for (int w = 0; w < WAVES_PER_BLOCK; ++w) s += s_wave[w];
        partials[blockIdx.x] = s;
    }
}

// ---------------------------------------------------------------------------
// Kernel 3: fixed-order final reduction -> scalar loss
// loss = codebook + BETA*commitment = (1 + 0.25) * mean((q-x)^2)
// ---------------------------------------------------------------------------
__global__ void vq_finalize_kernel(const float* __restrict__ partials,
                                   float* __restrict__ out_loss) {
    if (threadIdx.x == 0 && blockIdx.x == 0) {
        double s = 0.0;
        for (int i = 0; i < NUM_BLOCKS; ++i) s += (double)partials[i];
        out_loss[0] = (float)(1.25 * (s / (double)OUT_ELEMS));
    }
}

extern "C" void kernel_launch(void* const* d_in, const int* in_sizes, int n_in,
                              void* d_out, int out_size, void* d_ws, size_t ws_size,
                              hipStream_t stream) {
    const float* X = (const float*)d_in[0];   // 16*64*64*64 = 4194304 f32
    const float* E = (const float*)d_in[1];   // 1024*64     = 65536  f32
    float* out = (float*)d_out;               // 4194304 quantized + 1 loss
    float* ws  = (float*)d_ws;
    float* partials = ws;                     // [NUM_BLOCKS]
    float* enorm    = ws + 1024;              // [K_CODES]

    vq_enorm_kernel<<<(K_CODES + 255) / 256, 256, 0, stream>>>(E, enorm);
    vq_main_kernel<<<NUM_BLOCKS, BLOCK_THREADS, 0, stream>>>(
        X, E, enorm, out, partials);
    vq_finalize_kernel<<<1, 1, 0, stream>>>(partials, out + OUT_ELEMS);
}